// DimwiseMedianConv_2422361555340
// MI455X (gfx1250) — compile-verified
//
#include <hip/hip_runtime.h>

typedef __attribute__((ext_vector_type(2))) float v2f;
typedef __attribute__((ext_vector_type(8))) float v8f;

#define NNODES 10000
#define DIN    512
#define DOUT   256
#define KNBR   16
#define NVALS  (KNBR + 1)   // neighbors + self loop

// -------------------------------------------------------------------------
// Phase 1: h = feat(10000x512) @ weight(512x256), fp32, via V_WMMA_F32_16X16X4_F32.
// One wave32 computes one 16x16 output tile; 8 waves/block share the A slab.
// A fragment (16x4 f32, 2 VGPRs): lanes 0-15 hold row M=lane, K={0,1};
//                                 lanes 16-31 hold row M=lane-16, K={2,3}.
// C/D fragment (16x16 f32, 8 VGPRs): VGPR r = row r (lanes 0-15) / r+8 (16-31).
// -------------------------------------------------------------------------
__global__ __launch_bounds__(256) void dmc_gemm_wmma_f32(
    const float* __restrict__ feat,
    const float* __restrict__ weight,
    float* __restrict__ h) {
  const int lane   = threadIdx.x;         // 0..31 within wave
  const int laneLo = lane & 15;
  const int laneHi = lane >> 4;
  const int nTile  = blockIdx.x * 8 + threadIdx.y;  // 0..15
  const int m0     = blockIdx.y * 16;
  const int n0     = nTile * 16;

  const float* aBase = feat   + (size_t)(m0 + laneLo) * DIN + 2 * laneHi;
  const float* bBase = weight + (size_t)(2 * laneHi) * DOUT + n0 + laneLo;

  v8f c = {};
  for (int k = 0; k < DIN; k += 4) {
    // A: two consecutive K values per lane -> 8-byte contiguous load
    v2f a = *(const v2f*)(aBase + k);
    // B (4x16 f32, 2 VGPRs): VGPR r holds K = r + 2*laneHi, N = laneLo
    v2f b;
    b.x = bBase[(size_t)k * DOUT];
    b.y = bBase[(size_t)k * DOUT + DOUT];
    c = __builtin_amdgcn_wmma_f32_16x16x4_f32(
        /*neg_a=*/false, a, /*neg_b=*/false, b,
        /*c_mod=*/(short)0, c, /*reuse_a=*/false, /*reuse_b=*/false);
  }

  float* outp = h + (size_t)(m0 + 8 * laneHi) * DOUT + n0 + laneLo;
#pragma unroll
  for (int r = 0; r < 8; ++r)
    outp[(size_t)r * DOUT] = c[r];
}

// -------------------------------------------------------------------------
// Phase 2: per-(node, dim) weighted median over 16 neighbors + self (w=1).
// Selection form (order-independent, matches sorted-cumsum semantics):
//   pick value v with  sum{w_j : v_j < v} < half  and  sum{w_j : v_j <= v} >= half.
// Block = 256 threads = one node (one thread per output dim); neighbor ids
// and edge weights staged once in LDS. h rows are gathered coalesced and hit L2.
// -------------------------------------------------------------------------
__global__ __launch_bounds__(256) void dmc_median(
    const float* __restrict__ h,
    const int*   __restrict__ nbr,
    const float* __restrict__ ew,
    const float* __restrict__ bias,
    float* __restrict__ out) {
  __shared__ int   s_idx[KNBR];
  __shared__ float s_w[KNBR];
  const int node = blockIdx.x;
  const int dim  = threadIdx.x;

  if (dim < KNBR) {
    s_idx[dim] = nbr[(size_t)node * KNBR + dim];
    s_w[dim]   = ew[(size_t)node * KNBR + dim];
  }
  __syncthreads();

  float v[NVALS], w[NVALS];
  float wtot = 1.0f;  // self-loop weight
#pragma unroll
  for (int j = 0; j < KNBR; ++j) {
    v[j] = h[(size_t)s_idx[j] * DOUT + dim];
    w[j] = s_w[j];
    wtot += w[j];
  }
  v[KNBR] = h[(size_t)node * DOUT + dim];
  w[KNBR] = 1.0f;

  const float half = 0.5f * wtot;
  float best = v[KNBR];
#pragma unroll
  for (int i = 0; i < NVALS; ++i) {
    float below = 0.0f, upto = 0.0f;
#pragma unroll
    for (int j = 0; j < NVALS; ++j) {
      below += (v[j] <  v[i]) ? w[j] : 0.0f;
      upto  += (v[j] <= v[i]) ? w[j] : 0.0f;
    }
    if (below < half && upto >= half) best = v[i];
  }
  out[(size_t)node * DOUT + dim] = best + bias[dim];
}

// -------------------------------------------------------------------------
extern "C" void kernel_launch(void* const* d_in, const int* in_sizes, int n_in,
                              void* d_out, int out_size, void* d_ws, size_t ws_size,
                              hipStream_t stream) {
  (void)in_sizes; (void)n_in; (void)out_size; (void)ws_size;
  const float* feat   = (const float*)d_in[0];  // [N, DIN]
  const int*   nbr    = (const int*)  d_in[1];  // [N, K]
  const float* ew     = (const float*)d_in[2];  // [N, K]
  const float* weight = (const float*)d_in[3];  // [DIN, DOUT]
  const float* bias   = (const float*)d_in[4];  // [DOUT]
  float*       out    = (float*)d_out;          // [N, DOUT]
  float*       h      = (float*)d_ws;           // [N, DOUT] fp32 scratch (10.24 MB)

  // GEMM: grid (DOUT/16/8 = 2, N/16 = 625), block (32, 8) = 8 waves
  dim3 gblk(32, 8, 1);
  dim3 ggrid(DOUT / 16 / 8, NNODES / 16, 1);
  dmc_gemm_wmma_f32<<<ggrid, gblk, 0, stream>>>(feat, weight, h);

  // Median: one block per node, one thread per output dim
  dmc_median<<<NNODES, DOUT, 0, stream>>>(h, nbr, ew, bias, out);
}